// GroupEmbedding_85229331022140
// MI455X (gfx1250) — compile-verified
//
#include <hip/hip_runtime.h>

typedef float v2f __attribute__((ext_vector_type(2)));
typedef float v8f __attribute__((ext_vector_type(8)));

#define NF 128
#define G  16
#define F  8
#define D  512

// One wave computes a 16(batch) x 16(d) fp32 tile of out[:, g, :] using two
// chained V_WMMA_F32_16X16X4_F32 (K = 8 = 2 x 4). Block = 128 threads = 4
// waves; waves split D (32 tiles of 16) 8 tiles each. Grid = (B/16, G).
__global__ __launch_bounds__(128) void group_embed_wmma(
    const float* __restrict__ x,        // (B, NF)
    const float* __restrict__ Wm,       // (G, F, D)
    const float* __restrict__ bias,     // (G, D)
    const int*   __restrict__ gidx,     // (G, F)
    const int*   __restrict__ mgidx,    // (B,)
    float*       __restrict__ out)      // (B, G, D)
{
    const int lane  = threadIdx.x & 31;
    const int wave  = threadIdx.x >> 5;       // 0..3
    const int btile = blockIdx.x;             // 0..B/16-1
    const int g     = blockIdx.y;             // 0..G-1

    const int half  = lane >> 4;              // 0: lanes 0-15, 1: lanes 16-31
    const int lm    = lane & 15;
    const int kb    = half * 2;               // A/B VGPR0 holds K=kb, VGPR1 K=kb+1

    // ---- A operands (x gathered through group_idx), loaded once per wave ----
    const int* gi   = gidx + g * F;
    const int  f00  = gi[kb],     f01 = gi[kb + 1];
    const int  f10  = gi[kb + 4], f11 = gi[kb + 5];
    const int  row  = btile * 16 + lm;        // A-matrix row for this lane
    const float* xr = x + (long)row * NF;
    v2f a0; a0.x = xr[f00]; a0.y = xr[f01];   // K = kb, kb+1
    v2f a1; a1.x = xr[f10]; a1.y = xr[f11];   // K = kb+4, kb+5

    // ---- per-row zero mask: C/D VGPR r maps to row (r + half*8) of the tile ----
    float rmask[8];
#pragma unroll
    for (int r = 0; r < 8; ++r) {
        const int rr = btile * 16 + r + half * 8;
        rmask[r] = (mgidx[rr] == g) ? 0.0f : 1.0f;
    }

    const float* Wg = Wm + (long)g * F * D;   // (F, D) slab for this group
    const float* bg = bias + (long)g * D;

#pragma unroll
    for (int t = 0; t < 8; ++t) {
        const int dcol = (wave * 8 + t) * 16 + lm;   // this lane's N column in D

        // ---- B operands: rows K=kb..kb+1 (+4) of W[g], column dcol ----
        v2f b0; b0.x = Wg[(kb    ) * D + dcol]; b0.y = Wg[(kb + 1) * D + dcol];
        v2f b1; b1.x = Wg[(kb + 4) * D + dcol]; b1.y = Wg[(kb + 5) * D + dcol];

        v8f c = {};
        c = __builtin_amdgcn_wmma_f32_16x16x4_f32(false, a0, false, b0,
                                                  (short)0, c, false, false);
        c = __builtin_amdgcn_wmma_f32_16x16x4_f32(false, a1, false, b1,
                                                  (short)0, c, false, false);

        const float bv = bg[dcol];

        // ---- masked bias-add + non-temporal store (256 MB stream; keep L2 clean) ----
#pragma unroll
        for (int r = 0; r < 8; ++r) {
            const int rr = btile * 16 + r + half * 8;        // global batch row
            const float v = (c[r] + bv) * rmask[r];
            __builtin_nontemporal_store(v, out + ((long)rr * G + g) * D + dcol);
        }
    }
}

extern "C" void kernel_launch(void* const* d_in, const int* in_sizes, int n_in,
                              void* d_out, int out_size, void* d_ws, size_t ws_size,
                              hipStream_t stream) {
    const float* x     = (const float*)d_in[0];
    const float* Wm    = (const float*)d_in[1];
    const float* bias  = (const float*)d_in[2];
    const int*   gidx  = (const int*)d_in[3];
    const int*   mgidx = (const int*)d_in[4];
    float*       out   = (float*)d_out;

    const int Bn = in_sizes[0] / NF;          // 8192
    dim3 grid(Bn / 16, G);                    // (512, 16)
    group_embed_wmma<<<grid, 128, 0, stream>>>(x, Wm, bias, gidx, mgidx, out);
}